// RegionProposalNetwork_67765993996339
// MI455X (gfx1250) — compile-verified
//
#include <hip/hip_runtime.h>
#include <hip/hip_bf16.h>
#include <stdint.h>
#include <math.h>

// ---------------- tunables ----------------
#define BINS   2048      // histogram bins = top 11 bits of order-preserving key
#define CAP    8192      // max candidates captured above threshold (per batch)
#define MSEL   4096      // sorted candidates fed to greedy NMS (per batch)
#define ELEMS  32        // elements per thread in score pass
#define LOG_MAX_C 4.135166556742356f   // log(1000/16)
#define NMS_IOU 0.7f

typedef uint32_t u32x4 __attribute__((ext_vector_type(4)));
typedef uint32_t u32x8 __attribute__((ext_vector_type(8)));

// Order-preserving float->uint key (bigger key == bigger float).
__device__ __forceinline__ uint32_t f2key(float f) {
  uint32_t u = __float_as_uint(f);
  return (u & 0x80000000u) ? ~u : (u | 0x80000000u);
}

__device__ __forceinline__ float4 decode_clip(float4 a, float4 d, float imw, float imh) {
  float aw  = a.z - a.x, ah = a.w - a.y;
  float acx = a.x + 0.5f * aw, acy = a.y + 0.5f * ah;
  float dw  = fminf(d.z, LOG_MAX_C), dh = fminf(d.w, LOG_MAX_C);
  float pcx = fmaf(d.x, aw, acx), pcy = fmaf(d.y, ah, acy);
  float pw  = expf(dw) * aw, ph = expf(dh) * ah;
  float4 r;
  r.x = fminf(fmaxf(pcx - 0.5f * pw, 0.0f), imw);
  r.y = fminf(fmaxf(pcy - 0.5f * ph, 0.0f), imh);
  r.z = fminf(fmaxf(pcx + 0.5f * pw, 0.0f), imw);
  r.w = fminf(fmaxf(pcy + 0.5f * ph, 0.0f), imh);
  return r;
}

// TDM: one-descriptor 1D tile load (contiguous `bytes` from gptr -> LDS offset).
// D# layout per CDNA5 ISA 8.3/8.4: count=1, type=2("image"), data_size=3 (8B),
// single row: tensor_dim0 = tile_dim0 = bytes/8 (<= 65535), tensor_dim1 = tile_dim1 = 1.
__device__ __forceinline__ void tdm_load_1d(uint32_t lds_off, const void* gptr, uint32_t bytes) {
  uint32_t ne = bytes >> 3;                      // element count in 8B units
  uint64_t ga = (uint64_t)(uintptr_t)gptr;
  u32x4 g0;
  g0[0] = 1u;                                    // count=1, is_restore=0, gather off
  g0[1] = lds_off;                               // lds_addr (bytes)
  g0[2] = (uint32_t)ga;                          // global_addr[31:0]  (bits 95:64)
  g0[3] = (uint32_t)((ga >> 32) & 0x01FFFFFFu)   // global_addr[56:32] (bits 120:96)
        | (2u << 30);                            // type=2 (bits 127:126)
  u32x8 g1;
  g1[0] = (3u << 16);                            // wg_mask=0, data_size=3 (8B)
  g1[1] = (ne & 0xFFFFu) << 16;                  // tensor_dim0[15:0] @ bits 63:48
  g1[2] = ((ne >> 16) & 0xFFFFu) | (1u << 16);   // tensor_dim0[31:16] + tensor_dim1=1
  g1[3] = (ne & 0xFFFFu) << 16;                  // tile_dim0 @ bits 127:112
  g1[4] = 1u;                                    // tile_dim1=1, tile_dim2 unused
  g1[5] = ne;                                    // tensor_dim0_stride[31:0]
  g1[6] = (ne & 0xFFFFu) << 16;                  // tensor_dim1_stride[15:0] @ bits 223:208
  g1[7] = 0u;
  asm volatile("tensor_load_to_lds %0, %1" :: "s"(g0), "s"(g1) : "memory");
}

// -------- kernel 0: zero the per-launch counters --------
__global__ void k_zero(uint32_t* __restrict__ ghist, uint32_t* __restrict__ candCount, int B) {
  int i = blockIdx.x * blockDim.x + threadIdx.x;
  if (i < B * BINS) ghist[i] = 0u;
  if (i < B)        candCount[i] = 0u;
}

// -------- kernel 1: decode -> masked score + key histogram --------
__global__ void k_score_hist(const float4* __restrict__ anchors,
                             const float4* __restrict__ deltas,
                             const float*  __restrict__ obj,
                             const int* __restrict__ imh_p, const int* __restrict__ imw_p,
                             float* __restrict__ scores, uint32_t* __restrict__ ghist,
                             int N) {
  __shared__ uint32_t lh[BINS];
  const int b = blockIdx.y;
  for (int i = threadIdx.x; i < BINS; i += blockDim.x) lh[i] = 0u;
  __syncthreads();
  const float imw = (float)imw_p[0], imh = (float)imh_p[0];
  const int base = blockIdx.x * (blockDim.x * ELEMS);
  for (int e = 0; e < ELEMS; ++e) {
    int n = base + e * blockDim.x + threadIdx.x;
    if (n < N) {
      float4 a  = anchors[n];
      float4 d  = deltas[(size_t)b * N + n];
      float4 bx = decode_clip(a, d, imw, imh);
      float area = (bx.z - bx.x) * (bx.w - bx.y);
      float s = (area > 1.0f) ? obj[(size_t)b * N + n] : -INFINITY;
      scores[(size_t)b * N + n] = s;
      atomicAdd(&lh[f2key(s) >> 21], 1u);
    }
  }
  __syncthreads();
  for (int i = threadIdx.x; i < BINS; i += blockDim.x) {
    uint32_t c = lh[i];
    if (c) atomicAdd(&ghist[(size_t)b * BINS + i], c);
  }
}

// -------- kernel 2: find per-batch key threshold for top-MSEL --------
__global__ void k_threshold(const uint32_t* __restrict__ ghist, uint32_t* __restrict__ thrKey, int B) {
  int b = threadIdx.x;
  if (b >= B) return;
  uint32_t cum = 0; int T = 0;
  for (int bin = BINS - 1; bin >= 0; --bin) {
    uint32_t c = ghist[b * BINS + bin];
    if (cum + c >= MSEL) {
      // Keep the captured set deterministic: if including this bin would
      // overflow CAP (nondeterministic atomic-append drops), step up one bin.
      T = (cum + c <= CAP || cum == 0) ? bin : bin + 1;
      if (T > BINS - 1) T = BINS - 1;
      break;
    }
    cum += c;
  }
  thrKey[b] = ((uint32_t)T) << 21;   // accept all keys >= floor of threshold bin
}

// -------- kernel 3: compact indices above threshold --------
__global__ void k_compact(const float* __restrict__ scores, const uint32_t* __restrict__ thrKey,
                          uint32_t* __restrict__ candCount,
                          uint32_t* __restrict__ candKey, uint32_t* __restrict__ candIdx, int N) {
  int b = blockIdx.y;
  int n = blockIdx.x * blockDim.x + threadIdx.x;
  if (n >= N) return;
  uint32_t key = f2key(scores[(size_t)b * N + n]);
  if (key >= thrKey[b]) {
    uint32_t pos = atomicAdd(&candCount[b], 1u);
    if (pos < CAP) { candKey[b * CAP + pos] = key; candIdx[b * CAP + pos] = (uint32_t)n; }
  }
}

// -------- kernel 4: bitonic sort (key desc, idx asc) fully in LDS; gather top-MSEL --------
__global__ __launch_bounds__(1024) void k_sort_gather(
    const uint32_t* __restrict__ candCount, const uint32_t* __restrict__ candKey,
    const uint32_t* __restrict__ candIdx,
    const float4* __restrict__ anchors, const float4* __restrict__ deltas,
    const float* __restrict__ scores,
    const int* __restrict__ imh_p, const int* __restrict__ imw_p,
    float4* __restrict__ candBox, float* __restrict__ candScore, int N) {
  __shared__ uint64_t buf[CAP];            // 64 KB — fits easily in CDNA5's 320 KB WGP LDS
  const int b = blockIdx.x;
  const int tid = threadIdx.x;
  uint32_t cnt = candCount[b]; if (cnt > CAP) cnt = CAP;
  for (int s = tid; s < CAP; s += 1024)
    buf[s] = (s < (int)cnt)
      ? (((uint64_t)candKey[b * CAP + s] << 32) | (uint32_t)(~candIdx[b * CAP + s]))
      : 0ull;                              // worst possible composite -> sinks to the end
  for (unsigned k = 2; k <= CAP; k <<= 1) {
    for (unsigned j = k >> 1; j > 0; j >>= 1) {
      __syncthreads();
      for (unsigned i = tid; i < CAP; i += 1024) {
        unsigned ixj = i ^ j;
        if (ixj > i) {
          uint64_t x = buf[i], y = buf[ixj];
          bool desc = ((i & k) == 0);
          if (desc ? (x < y) : (x > y)) { buf[i] = y; buf[ixj] = x; }
        }
      }
    }
  }
  __syncthreads();
  const float imw = (float)imw_p[0], imh = (float)imh_p[0];
  for (int s = tid; s < MSEL; s += 1024) {
    uint64_t v = buf[s];
    float4 bx = make_float4(0.f, 0.f, 0.f, 0.f);
    float  sc = -INFINITY;
    if (v != 0ull) {
      uint32_t idx = ~((uint32_t)(v & 0xFFFFFFFFull));
      bx = decode_clip(anchors[idx], deltas[(size_t)b * N + idx], imw, imh);
      sc = scores[(size_t)b * N + idx];
    }
    candBox[b * MSEL + s]   = bx;
    candScore[b * MSEL + s] = sc;
  }
}

// -------- kernel 5: greedy NMS; boxes via TDM, scores via per-lane async->LDS --------
__global__ __launch_bounds__(1024) void k_nms(
    const float4* __restrict__ candBox, const float* __restrict__ candScore,
    float* __restrict__ out, int K) {
  __shared__ float4   sBox[MSEL];          // 64 KB
  __shared__ float    sScore[MSEL];        // 16 KB
  __shared__ uint32_t sSup[MSEL / 32];
  const int b = blockIdx.x;
  const int tid = threadIdx.x;

  // 64 KB box tile: single TDM descriptor, issued by wave 0 (TDM ignores EXEC).
  if (tid == 0) {
    tdm_load_1d((uint32_t)(uintptr_t)&sBox[0], &candBox[b * MSEL],
                (uint32_t)(MSEL * sizeof(float4)));
  }
  // 16 KB score tile: per-lane async global->LDS (ASYNCcnt path).
  {
    uint32_t lds = (uint32_t)(uintptr_t)&sScore[tid * 4];
    const float* g = &candScore[b * MSEL + tid * 4];
    asm volatile("global_load_async_to_lds_b128 %0, %1, off" :: "v"(lds), "v"(g) : "memory");
  }
  asm volatile("s_wait_asynccnt 0" ::: "memory");
  if (tid == 0) __builtin_amdgcn_s_wait_tensorcnt(0);   // issuing wave waits TDM
  for (int s = tid; s < MSEL / 32; s += 1024) sSup[s] = 0u;
  __syncthreads();                                      // publish LDS to all waves

  // Greedy pass. Suppression bits are written ONLY in keeper iterations, each of
  // which ends with a barrier — so dead-skip iterations need no barrier at all,
  // and the kept count is a uniform per-thread register.
  int kc = 0;
  for (int i = 0; i < MSEL; ++i) {
    if (kc >= K) break;                       // uniform
    float sc = sScore[i];
    if (!(sc > -INFINITY)) break;             // sorted: everything after is invalid
    if ((sSup[i >> 5] >> (i & 31)) & 1u) continue;   // stable since last barrier
    float4 bi = sBox[i];
    float  ai = (bi.z - bi.x) * (bi.w - bi.y);
    if (tid == 0) {
      float* o = out + ((size_t)b * K + kc) * 6;
      o[0] = bi.x; o[1] = bi.y; o[2] = bi.z; o[3] = bi.w; o[4] = sc; o[5] = 1.0f;
    }
    ++kc;
    for (int j = i + 1 + tid; j < MSEL; j += 1024) {
      float4 bj = sBox[j];
      float iw = fminf(bi.z, bj.z) - fmaxf(bi.x, bj.x);
      float ih = fminf(bi.w, bj.w) - fmaxf(bi.y, bj.y);
      float inter = fmaxf(iw, 0.0f) * fmaxf(ih, 0.0f);
      float aj = (bj.z - bj.x) * (bj.w - bj.y);
      float iou = inter / (ai + aj - inter + 1e-9f);
      if (iou > NMS_IOU) atomicOr(&sSup[j >> 5], 1u << (j & 31));
    }
    __syncthreads();                          // publish this keeper's suppression
  }
  for (int r = kc + tid; r < K; r += 1024) {  // zero-pad invalid rows
    float* o = out + ((size_t)b * K + r) * 6;
#pragma unroll
    for (int c = 0; c < 6; ++c) o[c] = 0.0f;
  }
}

extern "C" void kernel_launch(void* const* d_in, const int* in_sizes, int n_in,
                              void* d_out, int out_size, void* d_ws, size_t ws_size,
                              hipStream_t stream) {
  (void)n_in; (void)ws_size;
  const float4* anchors = (const float4*)d_in[0];
  const float4* deltas  = (const float4*)d_in[1];
  const float*  obj     = (const float*)d_in[2];
  const int*    imh     = (const int*)d_in[3];
  const int*    imw     = (const int*)d_in[4];
  (void)d_in[5]; // post_nms_top_n: derived from out_size instead (host-side, deterministic)

  const int N = in_sizes[0] / 4;
  const int B = in_sizes[2] / N;
  const int K = out_size / (B * 6);

  char* ws = (char*)d_ws;
  size_t off = 0;
  float4*   candBox   = (float4*)(ws + off);   off += (size_t)B * MSEL * sizeof(float4);
  float*    scores    = (float*)(ws + off);    off += (size_t)B * N * sizeof(float);
  float*    candScore = (float*)(ws + off);    off += (size_t)B * MSEL * sizeof(float);
  uint32_t* candKey   = (uint32_t*)(ws + off); off += (size_t)B * CAP * sizeof(uint32_t);
  uint32_t* candIdx   = (uint32_t*)(ws + off); off += (size_t)B * CAP * sizeof(uint32_t);
  uint32_t* ghist     = (uint32_t*)(ws + off); off += (size_t)B * BINS * sizeof(uint32_t);
  uint32_t* thrKey    = (uint32_t*)(ws + off); off += (size_t)B * sizeof(uint32_t);
  uint32_t* candCount = (uint32_t*)(ws + off); off += (size_t)B * sizeof(uint32_t);

  k_zero<<<(B * BINS + 255) / 256, 256, 0, stream>>>(ghist, candCount, B);

  dim3 gA((N + 256 * ELEMS - 1) / (256 * ELEMS), B);
  k_score_hist<<<gA, 256, 0, stream>>>(anchors, deltas, obj, imh, imw, scores, ghist, N);

  k_threshold<<<1, 64, 0, stream>>>(ghist, thrKey, B);

  dim3 gC((N + 255) / 256, B);
  k_compact<<<gC, 256, 0, stream>>>(scores, thrKey, candCount, candKey, candIdx, N);

  k_sort_gather<<<B, 1024, 0, stream>>>(candCount, candKey, candIdx, anchors, deltas,
                                        scores, imh, imw, candBox, candScore, N);

  k_nms<<<B, 1024, 0, stream>>>(candBox, candScore, (float*)d_out, K);
}